// GlobalAttentionLayer_71253507441407
// MI455X (gfx1250) — compile-verified
//
#include <hip/hip_runtime.h>
#include <hip/hip_bf16.h>
#include <math.h>

typedef __attribute__((ext_vector_type(2))) float v2f;
typedef __attribute__((ext_vector_type(8))) float v8f;

#define D_DIM 256
#define B_SEG 64

// order-preserving float <-> uint map (for atomic max on floats incl. negatives)
__device__ __forceinline__ unsigned flip_f32(float f) {
    unsigned u = __float_as_uint(f);
    return (u & 0x80000000u) ? ~u : (u | 0x80000000u);
}
__device__ __forceinline__ float unflip_f32(unsigned u) {
    unsigned v = (u & 0x80000000u) ? (u & 0x7FFFFFFFu) : ~u;
    return __uint_as_float(v);
}

__global__ void ga_init_kernel(unsigned* __restrict__ seg_max_u,
                               float* __restrict__ sum_e,
                               float* __restrict__ sum_eo) {
    int t = threadIdx.x;
    if (t < B_SEG) { seg_max_u[t] = 0u; sum_e[t] = 0.0f; }
    if (t < 2 * B_SEG) sum_eo[t] = 0.0f;
}

// Pass 1: fused [N,256] x [256,16] projection via V_WMMA_F32_16X16X4_F32.
// Columns: 0 = gate (states@Wg), 1..2 = out (states@Wo), 3..15 = zero pad.
// One wave per 16-row tile. Also does flip-encoded atomic seg-max of gate.
__global__ void ga_proj_kernel(const float* __restrict__ states,
                               const int* __restrict__ segids,
                               const float* __restrict__ Wg,
                               const float* __restrict__ bg,
                               const float* __restrict__ Wo,
                               float* __restrict__ gate_ws,
                               float* __restrict__ out_ws,
                               unsigned* __restrict__ seg_max_u,
                               int N, int ntiles) {
    __shared__ float w3[16 * D_DIM];  // transposed weights: w3[n*256 + k]

    const int t = threadIdx.x;  // 0..255 (8 waves)
    // Stage padded weight matrix into LDS (16 KB), transposed so each
    // B fragment is one ds_load_b64 of two consecutive-K values.
    for (int i = t; i < 16 * D_DIM; i += 256) {
        int n = i / D_DIM, k = i - n * D_DIM;
        float v = 0.0f;
        if (n == 0)      v = Wg[k];
        else if (n == 1) v = Wo[2 * k];
        else if (n == 2) v = Wo[2 * k + 1];
        w3[i] = v;
    }
    __syncthreads();

    const int lane = t & 31;
    const int wave = t >> 5;
    const int tile = blockIdx.x * 8 + wave;
    if (tile >= ntiles) return;  // wave-uniform: EXEC stays all-ones for WMMA

    const int mn = lane & 15;   // A row within tile / D column
    const int hi = lane >> 4;   // K-subgroup select
    const int row0 = tile * 16;

    int arow = row0 + mn;
    if (arow >= N) arow = N - 1;  // clamp pad rows (results discarded)
    const float* aptr = states + (size_t)arow * D_DIM + hi * 2;
    const float* bptr = &w3[mn * D_DIM + hi * 2];

    v8f c = {0.f, 0.f, 0.f, 0.f, 0.f, 0.f, 0.f, 0.f};
#pragma unroll 4
    for (int kc = 0; kc < D_DIM / 4; ++kc) {
        v2f a = *(const v2f*)(aptr + kc * 4);  // A[m][k0+2hi .. +1]
        v2f b = *(const v2f*)(bptr + kc * 4);  // B[k0+2hi .. +1][n]  (LDS)
        c = __builtin_amdgcn_wmma_f32_16x16x4_f32(
                /*neg_a=*/false, a, /*neg_b=*/false, b,
                /*c_mod=*/(short)0, c, /*reuse_a=*/false, /*reuse_b=*/false);
    }

    // D layout: lane holds column n = mn, rows r+8*hi in c[r]
    if (mn < 3) {
        const float bgv = bg[0];
#pragma unroll
        for (int r = 0; r < 8; ++r) {
            int row = row0 + r + 8 * hi;
            if (row < N) {
                float v = c[r];
                if (mn == 0) {
                    float g = v + bgv;
                    gate_ws[row] = g;
                    int s = segids[row];
                    atomicMax(&seg_max_u[s], flip_f32(g));
                } else {
                    out_ws[(size_t)row * 2 + (mn - 1)] = v;
                }
            }
        }
    }
}

// Pass 2: e = exp(gate - segmax); hierarchical segment reduce of
// sum(e) and sum(e*out) via LDS bins, then sparse global atomics.
__global__ void ga_reduce_kernel(const float* __restrict__ gate_ws,
                                 const float* __restrict__ out_ws,
                                 const int* __restrict__ segids,
                                 const unsigned* __restrict__ seg_max_u,
                                 float* __restrict__ sum_e,
                                 float* __restrict__ sum_eo, int N) {
    __shared__ float lds_e[B_SEG];
    __shared__ float lds_o[2 * B_SEG];
    const int t = threadIdx.x;
    if (t < B_SEG) lds_e[t] = 0.0f;
    if (t < 2 * B_SEG) lds_o[t] = 0.0f;
    __syncthreads();

    const int stride = gridDim.x * blockDim.x;
    for (int i = blockIdx.x * blockDim.x + t; i < N; i += stride) {
        int s = segids[i];
        float m = unflip_f32(seg_max_u[s]);
        float e = __expf(gate_ws[i] - m);
        atomicAdd(&lds_e[s], e);
        atomicAdd(&lds_o[2 * s],     e * out_ws[(size_t)2 * i]);
        atomicAdd(&lds_o[2 * s + 1], e * out_ws[(size_t)2 * i + 1]);
    }
    __syncthreads();

    if (t < B_SEG) {
        float v = lds_e[t];
        if (v != 0.0f) atomicAdd(&sum_e[t], v);
    } else if (t < 3 * B_SEG) {
        float v = lds_o[t - B_SEG];
        if (v != 0.0f) atomicAdd(&sum_eo[t - B_SEG], v);
    }
}

// Pass 3: pooled[b][c] = (sum_eo[b][c] + bo[c]*sum_e[b]) / (sum_e[b] + 1e-16)
__global__ void ga_final_kernel(const float* __restrict__ sum_e,
                                const float* __restrict__ sum_eo,
                                const float* __restrict__ bo,
                                float* __restrict__ out) {
    int t = threadIdx.x;
    if (t < 2 * B_SEG) {
        int b = t >> 1, cidx = t & 1;
        float se = sum_e[b];
        out[t] = (sum_eo[t] + bo[cidx] * se) / (se + 1e-16f);
    }
}

extern "C" void kernel_launch(void* const* d_in, const int* in_sizes, int n_in,
                              void* d_out, int out_size, void* d_ws, size_t ws_size,
                              hipStream_t stream) {
    const float* states = (const float*)d_in[0];
    const int*   segids = (const int*)d_in[1];
    const float* Wg     = (const float*)d_in[2];
    const float* bg     = (const float*)d_in[3];
    const float* Wo     = (const float*)d_in[4];
    const float* bo     = (const float*)d_in[5];
    float* out = (float*)d_out;

    const int N = in_sizes[1];  // number of rows (segment_ids length)

    // workspace: gate[N] | out[N*2] | seg_max_u[64] | sum_e[64] | sum_eo[128]
    float* wsf = (float*)d_ws;
    float*    gate_ws   = wsf;
    float*    out_ws    = wsf + (size_t)N;
    unsigned* seg_max_u = (unsigned*)(wsf + (size_t)3 * N);
    float*    sum_e     = wsf + (size_t)3 * N + B_SEG;
    float*    sum_eo    = wsf + (size_t)3 * N + 2 * B_SEG;

    ga_init_kernel<<<1, 128, 0, stream>>>(seg_max_u, sum_e, sum_eo);

    const int ntiles = (N + 15) / 16;
    const int blocksA = (ntiles + 7) / 8;  // 8 waves/block, 1 tile/wave
    ga_proj_kernel<<<blocksA, 256, 0, stream>>>(states, segids, Wg, bg, Wo,
                                                gate_ws, out_ws, seg_max_u,
                                                N, ntiles);

    ga_reduce_kernel<<<512, 256, 0, stream>>>(gate_ws, out_ws, segids,
                                              seg_max_u, sum_e, sum_eo, N);

    ga_final_kernel<<<1, 128, 0, stream>>>(sum_e, sum_eo, bo, out);
}